// PanoRenderer_40261023432758
// MI455X (gfx1250) — compile-verified
//
#include <hip/hip_runtime.h>
#include <math.h>

#define H_IMG 256
#define W_IMG 512
#define NRAYS (H_IMG * W_IMG)   // 131072
#define N_SAMP 300
#define N_CHUNK 75              // 300 / 4

typedef __attribute__((ext_vector_type(2))) float v2f;
typedef __attribute__((ext_vector_type(8))) float v8f;

// Volume after ground-concat is (65, 256, 256): plane iz==0 is constant 1000,
// planes 1..64 are the input voxel volume, zero padding outside.
// BRANCHLESS: always load from a clamped (legal) address, then select.
__device__ __forceinline__ float fetch_vox(const float* __restrict__ vox,
                                           int iz, int iy, int ix) {
    const bool valid = ((unsigned)iz <= 64u) & ((unsigned)iy <= 255u) &
                       ((unsigned)ix <= 255u);
    const int zi = min(max(iz - 1, 0), 63);
    const int yi = min(max(iy, 0), 255);
    const int xi = min(max(ix, 0), 255);
    float v = vox[(zi << 16) + (yi << 8) + xi];   // unconditional, always in-bounds
    v = (iz == 0) ? 1000.0f : v;                  // ground plane
    return valid ? v : 0.0f;                      // zero padding
}

// Trilinear sample; x,y,z already in voxel sample space (grid_sample applied).
__device__ __forceinline__ float trilinear(const float* __restrict__ vox,
                                           float x, float y, float z) {
    float x0f = floorf(x), y0f = floorf(y), z0f = floorf(z);
    float fx = x - x0f, fy = y - y0f, fz = z - z0f;
    int x0 = (int)x0f, y0 = (int)y0f, z0 = (int)z0f;
    float c000 = fetch_vox(vox, z0,     y0,     x0);
    float c001 = fetch_vox(vox, z0,     y0,     x0 + 1);
    float c010 = fetch_vox(vox, z0,     y0 + 1, x0);
    float c011 = fetch_vox(vox, z0,     y0 + 1, x0 + 1);
    float c100 = fetch_vox(vox, z0 + 1, y0,     x0);
    float c101 = fetch_vox(vox, z0 + 1, y0,     x0 + 1);
    float c110 = fetch_vox(vox, z0 + 1, y0 + 1, x0);
    float c111 = fetch_vox(vox, z0 + 1, y0 + 1, x0 + 1);
    float a0 = (1.0f - fx) * c000 + fx * c001;
    float a1 = (1.0f - fx) * c010 + fx * c011;
    float b0 = (1.0f - fx) * c100 + fx * c101;
    float b1 = (1.0f - fx) * c110 + fx * c111;
    float p0 = (1.0f - fy) * a0 + fy * a1;
    float p1 = (1.0f - fy) * b0 + fy * b1;
    return (1.0f - fz) * p0 + fz * p1;
}

__global__ void pano_init(int* mm) {
    mm[0] = 0x7F7FFFFF;  // +FLT_MAX bits (running min; depth >= 0)
    mm[1] = 0;           // 0.0f bits      (running max)
}

// One wave32 = 16 rays (M dim). Lanes 0-15 handle sample offsets {0,1} of each
// 4-sample chunk, lanes 16-31 offsets {2,3} -- exactly the 16x4 f32 A-matrix
// layout of V_WMMA_F32_16X16X4_F32. B col0 = depth values, col1 = ones.
// Accumulated D: col0 = depth per ray, col1 = opacity per ray.
__global__ __launch_bounds__(256) void pano_main(const float* __restrict__ vox,
                                                 float* __restrict__ out_depth,
                                                 float* __restrict__ out_opac,
                                                 int* __restrict__ mm) {
    const int lane    = threadIdx.x & 31;
    const int waveId  = blockIdx.x * (blockDim.x >> 5) + (threadIdx.x >> 5);
    const int rayBase = waveId << 4;
    const int m       = lane & 15;          // ray within the wave
    const int khi     = (lane >> 4) << 1;   // 0 for lanes<16, 2 for lanes>=16

    // Ray direction (pano): h = ray/512, w = ray%512
    const int ray = rayBase + m;
    const int hi  = ray >> 9;
    const int wj  = ray & 511;
    const float PI = 3.14159265358979323846f;
    float lat = (0.5f - (float)hi * (1.0f / 255.0f)) * PI;
    float lon = (-0.5f - 2.0f * (float)wj * (1.0f / 511.0f)) * PI;
    float cl = cosf(lat);
    float dx =  cl * cosf(lon);
    float dy = -cl * sinf(lon);
    float dz = sinf(lat);

    // Affine map t -> voxel sample coords:
    //   x = ((t*dx + 1)*256 - 1)/2           = t*(128*dx) + 127.5
    //   y = t*(128*dy) + 127.5
    //   z = ((2*(oz + t*dz) + 1 + 1)*65 - 1)/2 = t*(65*dz) + (65*oz + 64.5)
    const float oz = -1.0f + 4.0f / 128.0f;          // -0.96875
    const float Ax = 128.0f * dx;
    const float Ay = 128.0f * dy;
    const float Az = 65.0f * dz;
    const float Cz = 65.0f * oz + 64.5f;             // 1.53125
    const float intv = 64.0f / 300.0f;

    v8f acc = {};      // 16x16 f32 accumulator tile
    float S = 0.0f;    // running sum of sigma*intv for this lane's ray

    for (int c = 0; c < N_CHUNK; ++c) {
        const int   s0 = 4 * c + khi;                   // this lane's first sample
        const float t0 = (float)(s0 + 1) * (1.0f / 300.0f);
        const float t1 = (float)(s0 + 2) * (1.0f / 300.0f);

        float sg0 = trilinear(vox, t0 * Ax + 127.5f, t0 * Ay + 127.5f, t0 * Az + Cz);
        float sg1 = trilinear(vox, t1 * Ax + 127.5f, t1 * Ay + 127.5f, t1 * Az + Cz);
        float sd0 = sg0 * intv;
        float sd1 = sg1 * intv;

        // Cross half-wave prefix: lanes>=16 need the lower pair's sum first.
        float pair  = sd0 + sd1;
        float other = __shfl_xor(pair, 16, 32);
        float pre0  = S + ((lane >= 16) ? other : 0.0f);
        // prob_i = T_i * alpha_i; reuse E0*e0 = exp(-(pre0+sd0)) -> 3 exps / 2 samples
        float e0 = __expf(-sd0);
        float e1 = __expf(-sd1);
        float E0 = __expf(-pre0);
        float prob0 = E0 * (1.0f - e0);
        float prob1 = (E0 * e0) * (1.0f - e1);
        S += pair + other;

        v2f A; A.x = prob0; A.y = prob1;                 // 16x4 A: rows=rays, K split over halves
        float b0 = 0.0f, b1 = 0.0f;                      // 4x16 B: col0=depth_vals, col1=ones
        if (m == 0) {
            b0 = (float)(s0 + 1) * (64.0f / 300.0f);
            b1 = (float)(s0 + 2) * (64.0f / 300.0f);
        } else if (m == 1) {
            b0 = 1.0f; b1 = 1.0f;
        }
        v2f Bv; Bv.x = b0; Bv.y = b1;

        acc = __builtin_amdgcn_wmma_f32_16x16x4_f32(
            /*neg_a=*/false, A, /*neg_b=*/false, Bv,
            /*c_mod=*/(short)0, acc, /*reuse_a=*/false, /*reuse_b=*/false);
    }

    // D layout: lane n<16 VGPR g -> (M=g, N=n); lane n>=16 VGPR g -> (M=8+g, N=n-16).
    // Column N=0 (depth) lives in lanes 0/16; column N=1 (opacity) in lanes 1/17.
    if (m == 0) {
        float* outd = out_depth + rayBase + (khi << 2);  // +0 or +8
        float mn = 3.402823466e38f, mx = 0.0f;
        #pragma unroll
        for (int g = 0; g < 8; ++g) {
            float dv = acc[g];
            outd[g] = dv;
            mn = fminf(mn, dv);
            mx = fmaxf(mx, dv);
        }
        atomicMin(&mm[0], __float_as_int(mn));  // depth >= 0: int order == float order
        atomicMax(&mm[1], __float_as_int(mx));
    } else if (m == 1) {
        float* outo = out_opac + rayBase + (khi << 2);
        #pragma unroll
        for (int g = 0; g < 8; ++g) outo[g] = acc[g];
    }
}

__global__ void pano_norm(float* __restrict__ depth, const int* __restrict__ mm) {
    int idx = blockIdx.x * blockDim.x + threadIdx.x;
    if (idx >= NRAYS) return;
    float mn = __int_as_float(mm[0]);
    float mx = __int_as_float(mm[1]);
    float d = depth[idx];
    depth[idx] = (d - mn) / (mx - mn);
}

extern "C" void kernel_launch(void* const* d_in, const int* in_sizes, int n_in,
                              void* d_out, int out_size, void* d_ws, size_t ws_size,
                              hipStream_t stream) {
    const float* vox = (const float*)d_in[0];      // (1, 64, 256, 256) f32
    float* out_depth = (float*)d_out;              // first 131072 floats
    float* out_opac  = (float*)d_out + NRAYS;      // next 131072 floats
    int*   mm        = (int*)d_ws;                 // [min_bits, max_bits]

    pano_init<<<1, 1, 0, stream>>>(mm);
    // 8192 waves * 16 rays = 131072 rays; 8 waves (256 thr) per block -> 1024 blocks
    pano_main<<<NRAYS / 16 / 8, 256, 0, stream>>>(vox, out_depth, out_opac, mm);
    pano_norm<<<NRAYS / 256, 256, 0, stream>>>(out_depth, mm);
}